// HashGrid_29119878267164
// MI455X (gfx1250) — compile-verified
//
#include <hip/hip_runtime.h>

// Instant-NGP hash grid encode, MI455X (gfx1250, wave32).
// Gather-bound: 524288 pts x 16 levels x 8 corners = 67M independent 8B
// gathers from a 64MB table (fits in 192MB L2). One thread = one
// (point, level) pair; lane = point*16+level -> 256B/wave coalesced stores.

#define L_LEVELS   16
#define LOG_T      19
#define T_SIZE     (1u << LOG_T)       // 524288 entries per level
#define N_POINTS   524288
#define PRIME1     2654435761u
#define PRIME2     805459861u

typedef float v2f __attribute__((ext_vector_type(2)));

// CDNA5 async memory->LDS staging (no VGPR round-trip, ASYNCcnt-tracked).
// Guarded so the kernel compiles on toolchains lacking the builtins.
#if defined(__gfx1250__) &&                                        \
    __has_builtin(__builtin_amdgcn_global_load_async_to_lds_b32) && \
    __has_builtin(__builtin_amdgcn_s_wait_asynccnt)
#define USE_ASYNC_LDS 1
typedef __attribute__((address_space(1))) int* gptr_i32;
typedef __attribute__((address_space(3))) int* lptr_i32;
#else
#define USE_ASYNC_LDS 0
#endif

__global__ __launch_bounds__(256)
void HashGrid_29119878267164_kernel(const float* __restrict__ x,
                                    const float* __restrict__ hashmap,
                                    const float* __restrict__ resolution,
                                    float* __restrict__ out)
{
    __shared__ float s_xyz[48];   // 16 points * 3 coords
    __shared__ float s_res[L_LEVELS];

    const unsigned tid     = threadIdx.x;
    const unsigned blockPt = blockIdx.x * 16u;   // 16 points per 256-thread block

#if USE_ASYNC_LDS
    // Direct global->LDS async copies (global_load_async_to_lds_b32).
    if (tid < 48u) {
        __builtin_amdgcn_global_load_async_to_lds_b32(
            (gptr_i32)(x + (size_t)blockPt * 3u + tid),
            (lptr_i32)(&s_xyz[tid]),
            0, 0);
    }
    if (tid < L_LEVELS) {
        __builtin_amdgcn_global_load_async_to_lds_b32(
            (gptr_i32)(resolution + tid),
            (lptr_i32)(&s_res[tid]),
            0, 0);
    }
#else
    if (tid < 48u)      s_xyz[tid] = x[(size_t)blockPt * 3u + tid];
    if (tid < L_LEVELS) s_res[tid] = resolution[tid];
#endif

    // Pull next block's coordinate slice toward cache (global_prefetch_b8).
    if (tid == 0u && (blockPt + 16u) < N_POINTS) {
        __builtin_prefetch(&x[(size_t)(blockPt + 16u) * 3u], 0, 1);
    }

#if USE_ASYNC_LDS
    // Producer waves drain their async copies before the barrier; the
    // barrier then makes the LDS data visible to all 8 waves.
    __builtin_amdgcn_s_wait_asynccnt(0);
#endif
    __syncthreads();

    const unsigned ptLocal = tid >> 4;        // 0..15
    const unsigned lvl     = tid & 15u;       // 0..15
    const unsigned n       = blockPt + ptLocal;

    const float px  = s_xyz[ptLocal * 3u + 0u];
    const float py  = s_xyz[ptLocal * 3u + 1u];
    const float pz  = s_xyz[ptLocal * 3u + 2u];
    const float res = s_res[lvl];

    // Scaled coords, integer cell, trilinear weights. xn in [0,1) so
    // floor() is non-negative -> direct u32 convert (matches reference).
    const float sx = px * res, sy = py * res, sz = pz * res;
    const float fx = floorf(sx), fy = floorf(sy), fz = floorf(sz);
    const float wx = sx - fx,   wy = sy - fy,   wz = sz - fz;

    const unsigned cx0 = (unsigned)fx, cy0 = (unsigned)fy, cz0 = (unsigned)fz;
    const unsigned cx1 = cx0 + 1u,     cy1 = cy0 + 1u,     cz1 = cz0 + 1u;

    const unsigned hy0 = cy0 * PRIME1, hy1 = cy1 * PRIME1;
    const unsigned hz0 = cz0 * PRIME2, hz1 = cz1 * PRIME2;
    const unsigned m   = T_SIZE - 1u;

    // Corner order identical to reference: 000,001,010,011,100,101,110,111
    const unsigned i0 = (cx0 ^ hy0 ^ hz0) & m;
    const unsigned i1 = (cx0 ^ hy0 ^ hz1) & m;
    const unsigned i2 = (cx0 ^ hy1 ^ hz0) & m;
    const unsigned i3 = (cx0 ^ hy1 ^ hz1) & m;
    const unsigned i4 = (cx1 ^ hy0 ^ hz0) & m;
    const unsigned i5 = (cx1 ^ hy0 ^ hz1) & m;
    const unsigned i6 = (cx1 ^ hy1 ^ hz0) & m;
    const unsigned i7 = (cx1 ^ hy1 ^ hz1) & m;

    // Per-level table base; F=2 contiguous floats per entry -> b64 gathers.
    const v2f* __restrict__ tab = (const v2f*)hashmap + (size_t)lvl * T_SIZE;

    // Issue all 8 independent gathers back-to-back for max MLP.
    const v2f f0 = tab[i0];
    const v2f f1 = tab[i1];
    const v2f f2 = tab[i2];
    const v2f f3 = tab[i3];
    const v2f f4 = tab[i4];
    const v2f f5 = tab[i5];
    const v2f f6 = tab[i6];
    const v2f f7 = tab[i7];

    const float wxn = 1.0f - wx, wyn = 1.0f - wy, wzn = 1.0f - wz;
    const float w0 = wxn * wyn * wzn;
    const float w1 = wxn * wyn * wz;
    const float w2 = wxn * wy  * wzn;
    const float w3 = wxn * wy  * wz;
    const float w4 = wx  * wyn * wzn;
    const float w5 = wx  * wyn * wz;
    const float w6 = wx  * wy  * wzn;
    const float w7 = wx  * wy  * wz;

    v2f acc = f0 * w0;
    acc += f1 * w1;
    acc += f2 * w2;
    acc += f3 * w3;
    acc += f4 * w4;
    acc += f5 * w5;
    acc += f6 * w6;
    acc += f7 * w7;

    // Output [n, L*F] level-major; one contiguous 256B NT store per wave.
    // Non-temporal keeps the L2 reserved for the reused 64MB hash table.
    v2f* __restrict__ o = (v2f*)out + (size_t)n * L_LEVELS + lvl;
    __builtin_nontemporal_store(acc, o);
}

extern "C" void kernel_launch(void* const* d_in, const int* in_sizes, int n_in,
                              void* d_out, int out_size, void* d_ws, size_t ws_size,
                              hipStream_t stream) {
    const float* x          = (const float*)d_in[0];   // [N, 3]
    const float* hashmap    = (const float*)d_in[1];   // [L, T, F]
    const float* resolution = (const float*)d_in[2];   // [L]
    float* out              = (float*)d_out;           // [N, L*F]

    const int threads = 256;                   // 8 waves (wave32)
    const int blocks  = N_POINTS / 16;         // 16 points/block -> 32768
    HashGrid_29119878267164_kernel<<<blocks, threads, 0, stream>>>(
        x, hashmap, resolution, out);
}